// PointNetSetAbstraction_30683246363223
// MI455X (gfx1250) — compile-verified
//
#include <hip/hip_runtime.h>
#include <hip/hip_bf16.h>

typedef __attribute__((ext_vector_type(2))) float v2f;
typedef __attribute__((ext_vector_type(8))) float v8f;

#define BB    16
#define NN    4096
#define NPT   1024
#define NSMP  32
#define LL    (NPT*NSMP)   /* 32768 columns per batch */
#define INCH  64
#define TEDIM 128
#define R2C   0.04f
#define EPSV  1e-5f
#define NT    128          /* columns per GEMM block */
#define PSTR  (NT + 16)    /* K-pair row stride: +16 pairs => half-waves hit disjoint bank pairs */

__device__ __forceinline__ float gelu_f(float x) {
  return 0.5f * x * (1.0f + erff(x * 0.70710678118654752f));
}

// ---------------------------------------------------------------------------
// 1) Farthest point sampling: one block per batch, 1024 threads, 4 pts/thread.
// ---------------------------------------------------------------------------
__global__ __launch_bounds__(1024)
void fps_kernel(const float* __restrict__ xyz, float* __restrict__ newxyz,
                int* __restrict__ fpsidx) {
  const int b = blockIdx.x, tid = threadIdx.x;
  const float* X = xyz + (size_t)b * NN * 3;
  float px[4], py[4], pz[4], d[4];
#pragma unroll
  for (int i = 0; i < 4; ++i) {
    int n = tid + i * 1024;
    px[i] = X[n*3+0]; py[i] = X[n*3+1]; pz[i] = X[n*3+2];
    d[i] = 1e10f;
  }
  __shared__ float sval[1024];
  __shared__ int   sidx[1024];
  __shared__ float cen[3];
  __shared__ int   sfar;
  if (tid == 0) sfar = 0;
  __syncthreads();
  for (int it = 0; it < NPT; ++it) {
    int far = sfar;
    if (tid == 0) {
      float cx = X[far*3+0], cy = X[far*3+1], cz = X[far*3+2];
      cen[0] = cx; cen[1] = cy; cen[2] = cz;
      fpsidx[b*NPT + it] = far;
      newxyz[((size_t)b*NPT + it)*3 + 0] = cx;
      newxyz[((size_t)b*NPT + it)*3 + 1] = cy;
      newxyz[((size_t)b*NPT + it)*3 + 2] = cz;
    }
    __syncthreads();
    float cx = cen[0], cy = cen[1], cz = cen[2];
    float best = -1.0f; int bidx = 0x7fffffff;
#pragma unroll
    for (int i = 0; i < 4; ++i) {
      int n = tid + i * 1024;
      float dx = px[i]-cx, dy = py[i]-cy, dz = pz[i]-cz;
      float dd = dx*dx + dy*dy + dz*dz;
      d[i] = fminf(d[i], dd);
      if (d[i] > best || (d[i] == best && n < bidx)) { best = d[i]; bidx = n; }
    }
    sval[tid] = best; sidx[tid] = bidx;
    __syncthreads();
    for (int st = 512; st > 0; st >>= 1) {
      if (tid < st) {
        float v1 = sval[tid], v2 = sval[tid+st];
        int   i1 = sidx[tid], i2 = sidx[tid+st];
        if (v2 > v1 || (v2 == v1 && i2 < i1)) { sval[tid] = v2; sidx[tid] = i2; }
      }
      __syncthreads();
    }
    if (tid == 0) sfar = sidx[0];
    __syncthreads();
  }
}

// ---------------------------------------------------------------------------
// 2) Ball query: one wave32 per centroid; first NSMP in-range indices in
//    ascending index order (ballot + prefix popcount), pad with slot 0.
// ---------------------------------------------------------------------------
__global__ __launch_bounds__(256)
void ball_query(const float* __restrict__ xyz, const float* __restrict__ nxyz,
                int* __restrict__ idxout) {
  __shared__ int sl[8][NSMP];
  const int tid = threadIdx.x, wv = tid >> 5, lane = tid & 31;
  const int gw = blockIdx.x * 8 + wv;        // global centroid id
  const int b  = gw >> 10;
  const float* X = xyz + (size_t)b * NN * 3;
  const float cx = nxyz[(size_t)gw*3+0];
  const float cy = nxyz[(size_t)gw*3+1];
  const float cz = nxyz[(size_t)gw*3+2];
  int cnt = 0;
  for (int n0 = 0; n0 < NN && cnt < NSMP; n0 += 32) {
    int n = n0 + lane;
    float dx = X[n*3+0]-cx, dy = X[n*3+1]-cy, dz = X[n*3+2]-cz;
    float sq = dx*dx + dy*dy + dz*dz;
    bool within = sq <= R2C;
    unsigned mask = (unsigned)__ballot(within);
    int pos = cnt + __popc((int)(mask & ((1u << lane) - 1u)));
    if (within && pos < NSMP) sl[wv][pos] = n;
    cnt += __popc((int)mask);
  }
  __syncthreads();
  int cu = cnt < NSMP ? cnt : NSMP;
  int v = (lane < cu) ? sl[wv][lane] : ((cu > 0) ? sl[wv][0] : 0);
  idxout[(size_t)gw * NSMP + lane] = v;
}

// ---------------------------------------------------------------------------
// 3) Build X0 (B, 68, 32768): ch 0..2 = grouped_xyz - center, 3..66 = gathered
//    point features, ch 67 = zero pad (K multiple of 4 for WMMA).
// ---------------------------------------------------------------------------
__global__ __launch_bounds__(256)
void gather_build(const float* __restrict__ xyz, const float* __restrict__ points,
                  const float* __restrict__ nxyz, const int* __restrict__ idx,
                  float* __restrict__ X0) {
  const int tid = threadIdx.x, wv = tid >> 5, lane = tid & 31;
  const int gw = blockIdx.x * 8 + wv;
  const int b = gw >> 10, s = gw & 1023;
  const int id = idx[(size_t)gw * NSMP + lane];
  const float cx = nxyz[(size_t)gw*3+0];
  const float cy = nxyz[(size_t)gw*3+1];
  const float cz = nxyz[(size_t)gw*3+2];
  const float* X = xyz + (size_t)b * NN * 3;
  const float* P = points + (size_t)b * INCH * NN;
  float* dst = X0 + (size_t)b * 68 * LL;
  const size_t l = (size_t)s * NSMP + lane;
  dst[(size_t)0*LL + l] = X[id*3+0] - cx;
  dst[(size_t)1*LL + l] = X[id*3+1] - cy;
  dst[(size_t)2*LL + l] = X[id*3+2] - cz;
#pragma unroll 4
  for (int c = 0; c < INCH; ++c)
    dst[(size_t)(3+c)*LL + l] = P[(size_t)c*NN + id];
  dst[(size_t)67*LL + l] = 0.0f;
}

// ---------------------------------------------------------------------------
// 4) t-embedding bias: tb[b][o] = lb[o] + sum_j gelu(te[b][j]) * lw[o][j]
// ---------------------------------------------------------------------------
__global__ void tbias_kernel(const float* __restrict__ te, const float* __restrict__ lw,
                             const float* __restrict__ lb, float* __restrict__ tb,
                             int cout) {
  int i = blockIdx.x * 256 + threadIdx.x;
  if (i >= BB * cout) return;
  int b = i / cout, o = i % cout;
  float s = lb[o];
  for (int j = 0; j < TEDIM; ++j)
    s += gelu_f(te[b*TEDIM + j]) * lw[o*TEDIM + j];
  tb[i] = s;
}

// ---------------------------------------------------------------------------
// 5) WMMA f32 GEMM: dst[b][o][l] = sum_c W[o][c] * f(src[b][c][l]) + tb[b][o]
//    f = identity (layer 1) or fused batchnorm-normalize + GELU of the
//    previous layer (applied while staging the column strip into LDS).
//    LDS holds X as K-pairs: element (c,n) at ((c/2)*PSTR + n)*2 + (c&1), so
//    the B operand for a WMMA is one aligned ds_load_b64 (no repacking movs);
//    PSTR = NT+16 makes the two half-waves hit disjoint bank pairs.
// ---------------------------------------------------------------------------
template<int CINP, int CINW, int COUT, bool FUSE>
__global__ __launch_bounds__(256)
void gemm_wmma(const float* __restrict__ W, const float* __restrict__ src,
               const float* __restrict__ tb, float* __restrict__ dst,
               const float* __restrict__ stats, const float* __restrict__ gam,
               const float* __restrict__ bet) {
  constexpr int KST = CINP / 4;
  constexpr int MT = COUT / 16;
  constexpr int NGROUPS = 8 / MT;     // 2 for COUT=64, 1 for COUT=128
  constexpr int NSUB = 8 / NGROUPS;   // N subtiles per wave
  __shared__ v2f lds2[(CINP / 2) * PSTR];
  float* ldsf = (float*)lds2;
  const int tid = threadIdx.x;
  const int b = blockIdx.y;
  const int n0 = blockIdx.x * NT;

  // Stage strip of X into LDS in K-pair layout (fused norm+gelu for layers 2/3)
  for (int i = tid; i < CINP * NT; i += 256) {
    int c = i >> 7, n = i & (NT - 1);
    float v = src[((size_t)(b * CINP + c)) * LL + n0 + n];
    if (FUSE) {
      float m = stats[2*c], r = stats[2*c+1];
      v = gam[c] * (v - m) * r + bet[c];
      v = gelu_f(v);
    }
    ldsf[((c >> 1) * PSTR + n) * 2 + (c & 1)] = v;
  }
  __syncthreads();

  const int wv = tid >> 5, lane = tid & 31;
  const int mi = wv % MT, ng = wv / MT;
  const int lo = lane & 15;
  const int hi = lane >> 4;            // 0: K 0/1, rows M..  1: K 2/3, rows M+8..
  const int kofs = hi * 2;
  const int mrow = mi * 16 + lo;

  // A matrix (W) preload: VGPR0 = K+0 (lanes 0-15) / K+2 (lanes 16-31), VGPR1 = K+1 / K+3
  v2f a[KST];
#pragma unroll
  for (int ks = 0; ks < KST; ++ks) {
    int k = ks * 4 + kofs;
    a[ks].x = (k     < CINW) ? W[mrow * CINW + k]     : 0.0f;
    a[ks].y = (k + 1 < CINW) ? W[mrow * CINW + k + 1] : 0.0f;
  }

  const int rowbase = mi * 16 + hi * 8;  // C/D layout: VGPR r -> row r (+8 for hi lanes)
  float tbv[8];
#pragma unroll
  for (int r = 0; r < 8; ++r) tbv[r] = tb[b * COUT + rowbase + r];

  // B pointer for this lane: pair-row (2*ks + hi), column n
  const v2f* bp = lds2 + (size_t)hi * PSTR;

  for (int j = 0; j < NSUB; ++j) {
    const int n = (ng * NSUB + j) * 16 + lo;
    v8f acc = {};
#pragma unroll
    for (int ks = 0; ks < KST; ++ks) {
      v2f bv = bp[(2 * ks) * PSTR + n];   // one aligned ds_load_b64
      acc = __builtin_amdgcn_wmma_f32_16x16x4_f32(false, a[ks], false, bv,
                                                  (short)0, acc, false, false);
    }
    const size_t l = (size_t)n0 + n;
#pragma unroll
    for (int r = 0; r < 8; ++r)
      dst[((size_t)(b * COUT + rowbase + r)) * LL + l] = acc[r] + tbv[r];
  }
}

// ---------------------------------------------------------------------------
// 6) Deterministic two-stage per-channel mean/var reduction over (B, L)
// ---------------------------------------------------------------------------
template<int COUT>
__global__ __launch_bounds__(256)
void reduce_part(const float* __restrict__ Y, float* __restrict__ part) {
  const int ch = blockIdx.x, chunk = blockIdx.y, tid = threadIdx.x;
  const int b = chunk >> 2, l0 = (chunk & 3) * 8192;
  const float* p = Y + ((size_t)(b * COUT + ch)) * LL + l0;
  float s = 0.0f, q = 0.0f;
  for (int i = tid; i < 8192; i += 256) { float v = p[i]; s += v; q += v * v; }
  __shared__ float ss[256], sq[256];
  ss[tid] = s; sq[tid] = q;
  __syncthreads();
  for (int st = 128; st > 0; st >>= 1) {
    if (tid < st) { ss[tid] += ss[tid+st]; sq[tid] += sq[tid+st]; }
    __syncthreads();
  }
  if (tid == 0) {
    part[(ch * 64 + chunk) * 2 + 0] = ss[0];
    part[(ch * 64 + chunk) * 2 + 1] = sq[0];
  }
}

__global__ void reduce_final(const float* __restrict__ part, float* __restrict__ stats) {
  const int ch = blockIdx.x, tid = threadIdx.x;   // 64 threads
  __shared__ float ss[64], sq[64];
  ss[tid] = part[(ch * 64 + tid) * 2 + 0];
  sq[tid] = part[(ch * 64 + tid) * 2 + 1];
  __syncthreads();
  for (int st = 32; st > 0; st >>= 1) {
    if (tid < st) { ss[tid] += ss[tid+st]; sq[tid] += sq[tid+st]; }
    __syncthreads();
  }
  if (tid == 0) {
    const float inv = 1.0f / (float)((size_t)BB * LL);
    float m = ss[0] * inv;
    float var = sq[0] * inv - m * m;
    stats[ch * 2 + 0] = m;
    stats[ch * 2 + 1] = rsqrtf(var + EPSV);
  }
}

// ---------------------------------------------------------------------------
// 7) Final: normalize + GELU + max over K=32 -> (B, 128, 1024)
// ---------------------------------------------------------------------------
__global__ __launch_bounds__(256)
void norm_gelu_max(const float* __restrict__ Y, const float* __restrict__ stats,
                   const float* __restrict__ gam, const float* __restrict__ bet,
                   float* __restrict__ out) {
  const int gid = blockIdx.x * 256 + threadIdx.x;   // B*128*1024 threads
  const int s = gid & 1023;
  const int o = (gid >> 10) & 127;
  const int b = gid >> 17;
  const float m = stats[2*o], r = stats[2*o+1], g = gam[o], be = bet[o];
  const float4* p = (const float4*)(Y + ((size_t)(b * 128 + o)) * LL + (size_t)s * NSMP);
  float mx = -3.4e38f;
#pragma unroll
  for (int i = 0; i < 8; ++i) {
    float4 v = p[i];
    mx = fmaxf(mx, gelu_f(g * (v.x - m) * r + be));
    mx = fmaxf(mx, gelu_f(g * (v.y - m) * r + be));
    mx = fmaxf(mx, gelu_f(g * (v.z - m) * r + be));
    mx = fmaxf(mx, gelu_f(g * (v.w - m) * r + be));
  }
  out[((size_t)(b * 128 + o)) * NPT + s] = mx;
}

// ---------------------------------------------------------------------------
extern "C" void kernel_launch(void* const* d_in, const int* in_sizes, int n_in,
                              void* d_out, int out_size, void* d_ws, size_t ws_size,
                              hipStream_t stream) {
  (void)in_sizes; (void)n_in; (void)out_size; (void)ws_size;
  const float* xyz    = (const float*)d_in[0];
  const float* points = (const float*)d_in[1];
  const float* te     = (const float*)d_in[2];
  const float* W0  = (const float*)d_in[3];
  const float* lw0 = (const float*)d_in[4];
  const float* lb0 = (const float*)d_in[5];
  const float* g0  = (const float*)d_in[6];
  const float* be0 = (const float*)d_in[7];
  const float* W1  = (const float*)d_in[8];
  const float* lw1 = (const float*)d_in[9];
  const float* lb1 = (const float*)d_in[10];
  const float* g1  = (const float*)d_in[11];
  const float* be1 = (const float*)d_in[12];
  const float* W2  = (const float*)d_in[13];
  const float* lw2 = (const float*)d_in[14];
  const float* lb2 = (const float*)d_in[15];
  const float* g2  = (const float*)d_in[16];
  const float* be2 = (const float*)d_in[17];

  float* out_xyz = (float*)d_out;                       // (16, 1024, 3)
  float* out_pts = out_xyz + (size_t)BB * NPT * 3;      // (16, 128, 1024)

  char* w = (char*)d_ws;
  int*   fpsidx  = (int*)w;                             //  64 KB
  int*   ballidx = (int*)(w + (64u << 10));             //   2 MB
  float* tb0     = (float*)(w + (64u << 10) + (2u << 21));
  float* tb1     = tb0 + BB * 64;
  float* tb2     = tb1 + BB * 64;
  float* st0     = tb2 + BB * 128;
  float* st1     = st0 + 256;
  float* st2     = st1 + 256;
  float* part    = st2 + 256;                           // 128*64*2 floats
  float* X0      = (float*)(w + (4u << 20));            // (16, 68, 32768)
  float* Y0      = X0 + (size_t)BB * 68 * LL;           // (16, 64, 32768)
  float* Y1      = Y0 + (size_t)BB * 64 * LL;           // (16, 64, 32768)
  float* Y2      = Y1 + (size_t)BB * 64 * LL;           // (16, 128, 32768)

  // geometry
  fps_kernel<<<dim3(BB), dim3(1024), 0, stream>>>(xyz, out_xyz, fpsidx);
  ball_query<<<dim3(BB * NPT / 8), dim3(256), 0, stream>>>(xyz, out_xyz, ballidx);
  gather_build<<<dim3(BB * NPT / 8), dim3(256), 0, stream>>>(xyz, points, out_xyz, ballidx, X0);

  // conditioning biases
  tbias_kernel<<<dim3((BB * 64 + 255) / 256), dim3(256), 0, stream>>>(te, lw0, lb0, tb0, 64);
  tbias_kernel<<<dim3((BB * 64 + 255) / 256), dim3(256), 0, stream>>>(te, lw1, lb1, tb1, 64);
  tbias_kernel<<<dim3((BB * 128 + 255) / 256), dim3(256), 0, stream>>>(te, lw2, lb2, tb2, 128);

  // layer 1: 67->64 (X0 padded to 68), raw store for batchnorm stats
  gemm_wmma<68, 67, 64, false><<<dim3(LL / NT, BB), dim3(256), 0, stream>>>(
      W0, X0, tb0, Y0, nullptr, nullptr, nullptr);
  reduce_part<64><<<dim3(64, 64), dim3(256), 0, stream>>>(Y0, part);
  reduce_final<<<dim3(64), dim3(64), 0, stream>>>(part, st0);

  // layer 2: 64->64, norm+gelu of layer-1 fused into LDS staging
  gemm_wmma<64, 64, 64, true><<<dim3(LL / NT, BB), dim3(256), 0, stream>>>(
      W1, Y0, tb1, Y1, st0, g0, be0);
  reduce_part<64><<<dim3(64, 64), dim3(256), 0, stream>>>(Y1, part);
  reduce_final<<<dim3(64), dim3(64), 0, stream>>>(part, st1);

  // layer 3: 64->128
  gemm_wmma<64, 64, 128, true><<<dim3(LL / NT, BB), dim3(256), 0, stream>>>(
      W2, Y1, tb2, Y2, st1, g1, be1);
  reduce_part<128><<<dim3(128, 64), dim3(256), 0, stream>>>(Y2, part);
  reduce_final<<<dim3(128), dim3(64), 0, stream>>>(part, st2);

  // normalize + gelu + max over K
  norm_gelu_max<<<dim3((size_t)BB * 128 * NPT / 256), dim3(256), 0, stream>>>(
      Y2, st2, g2, be2, out_pts);
}